// SLADGNN_59983513256401
// MI455X (gfx1250) — compile-verified
//
#include <hip/hip_runtime.h>
#include <hip/hip_bf16.h>
#include <math.h>

typedef float v2f __attribute__((ext_vector_type(2)));
typedef float v8f __attribute__((ext_vector_type(8)));

// ---------------- degree / normalization ----------------

__global__ void deg_init_kernel(float* __restrict__ deg, int n) {
  int i = blockIdx.x * blockDim.x + threadIdx.x;
  if (i < n) deg[i] = 1.0f;  // self-loop contributes 1 to every node
}

__global__ void deg_acc_kernel(const int* __restrict__ dst, float* __restrict__ deg, int nE) {
  int e = blockIdx.x * blockDim.x + threadIdx.x;
  if (e < nE) atomicAdd(&deg[dst[e]], 1.0f);
}

__global__ void dinv_kernel(const float* __restrict__ deg, float* __restrict__ dinv, int n) {
  int i = blockIdx.x * blockDim.x + threadIdx.x;
  if (i < n) dinv[i] = rsqrtf(deg[i]);
}

// ---------------- WMMA fp32 GEMM: T[M,N] = A[M,K] @ W[K,N] ----------------
// One wave32 per 16-row tile; accumulates all N/16 column tiles in registers.
// Uses V_WMMA_F32_16X16X4_F32 (full fp32 precision on the matrix pipe).

template<int K, int N>
__global__ __launch_bounds__(256) void gcn_gemm_wmma(const float* __restrict__ A,
                                                     const float* __restrict__ W,
                                                     float* __restrict__ T, int M) {
  constexpr int NT = N / 16;
  const int wave = threadIdx.x >> 5;
  const int lane = threadIdx.x & 31;
  const int mtile = blockIdx.x * (blockDim.x >> 5) + wave;
  const int m0 = mtile * 16;
  if (m0 >= M) return;                       // wave-uniform exit (EXEC stays all-1s for WMMA)

  const int halfsel = lane >> 4;             // 0: lanes 0-15, 1: lanes 16-31
  const int laneM   = lane & 15;
  const int koff    = halfsel << 1;          // A/B VGPR layout: K=0,1 | K=2,3 split across halves

  v8f acc[NT] = {};

  for (int k0 = 0; k0 < K; k0 += 4) {
    // A fragment (16x4 fp32): lane holds A[m0+laneM, k0+koff .. +1]
    const float* ap = A + (size_t)(m0 + laneM) * K + k0 + koff;
    v2f a;
    a.x = ap[0];
    a.y = ap[1];
#pragma unroll
    for (int nt = 0; nt < NT; ++nt) {
      const int col = nt * 16 + laneM;
      // B fragment (4x16 fp32): lane holds W[k0+koff, col], W[k0+koff+1, col]
      v2f b;
      b.x = W[(size_t)(k0 + koff) * N + col];
      b.y = W[(size_t)(k0 + koff + 1) * N + col];
      acc[nt] = __builtin_amdgcn_wmma_f32_16x16x4_f32(
          /*neg_a=*/false, a, /*neg_b=*/false, b,
          /*c_mod=*/(short)0, acc[nt], /*reuse_a=*/false, /*reuse_b=*/false);
    }
  }

  // C/D layout: VGPR v -> row m0 + v + (halfsel ? 8 : 0), col = nt*16 + laneM
  const int rbase = m0 + (halfsel << 3);
#pragma unroll
  for (int nt = 0; nt < NT; ++nt) {
    const int col = nt * 16 + laneM;
#pragma unroll
    for (int v = 0; v < 8; ++v) {
      T[(size_t)(rbase + v) * N + col] = acc[nt][v];
    }
  }
}

// ---------------- edge gather/scale/scatter-add ----------------
// One wave per edge; lanes stripe features. dst accumulator lives in L2 (25.6MB << 192MB).

__global__ __launch_bounds__(256) void edge_agg_kernel(const int* __restrict__ src,
                                                       const int* __restrict__ dst,
                                                       const float* __restrict__ dinv,
                                                       const float* __restrict__ T,
                                                       float* __restrict__ H,
                                                       int nE, int N) {
  const int wave = threadIdx.x >> 5;
  const int lane = threadIdx.x & 31;
  const int e = blockIdx.x * (blockDim.x >> 5) + wave;
  if (e >= nE) return;
  const int s = src[e];
  const int d = dst[e];
  const float w = dinv[s] * dinv[d];
  const float* tp = T + (size_t)s * N;
  float* hp = H + (size_t)d * N;
  for (int f = lane; f < N; f += 32) {
    atomicAdd(&hp[f], tp[f] * w);
  }
}

// self-loop message + bias + ReLU (in place on the accumulator)
__global__ void finish_kernel(float* __restrict__ H, const float* __restrict__ T,
                              const float* __restrict__ dinv, const float* __restrict__ bias,
                              int M, int N) {
  int idx = blockIdx.x * blockDim.x + threadIdx.x;
  if (idx >= M * N) return;
  int i = idx / N;
  int f = idx - i * N;
  float di = dinv[i];
  float v = H[idx] + T[idx] * di * di + bias[f];
  H[idx] = v > 0.0f ? v : 0.0f;
}

// ---------------- prototype distances + MLP readout ----------------

__global__ __launch_bounds__(256) void readout_kernel(const float* __restrict__ H,
                                                      const float* __restrict__ prot,
                                                      const float* __restrict__ Wf0,
                                                      const float* __restrict__ bf0,
                                                      const float* __restrict__ Wf1,
                                                      const float* __restrict__ bf1,
                                                      const int* __restrict__ y,
                                                      float* __restrict__ out, int M) {
  __shared__ float sP[16 * 64];
  __shared__ float sPn[16];
  __shared__ float sW0[16 * 8];
  __shared__ float sb0[8];
  __shared__ float sW1[8];
  __shared__ float sb1;

  for (int t = threadIdx.x; t < 16 * 64; t += blockDim.x) sP[t] = prot[t];
  if (threadIdx.x < 128) sW0[threadIdx.x] = Wf0[threadIdx.x];
  if (threadIdx.x < 8) { sb0[threadIdx.x] = bf0[threadIdx.x]; sW1[threadIdx.x] = Wf1[threadIdx.x]; }
  if (threadIdx.x == 0) sb1 = bf1[0];
  __syncthreads();
  if (threadIdx.x < 16) {
    float s = 0.0f;
    for (int j = 0; j < 64; ++j) { float p = sP[threadIdx.x * 64 + j]; s += p * p; }
    sPn[threadIdx.x] = s;
  }
  __syncthreads();

  int i = blockIdx.x * blockDim.x + threadIdx.x;
  if (i >= M) return;

  float h[64];
  float hh = 0.0f;
  const float* hp = H + (size_t)i * 64;
#pragma unroll
  for (int j = 0; j < 64; ++j) { h[j] = hp[j]; hh += h[j] * h[j]; }

  float sim[16];
#pragma unroll
  for (int k = 0; k < 16; ++k) {
    float dot = 0.0f;
    for (int j = 0; j < 64; ++j) dot += h[j] * sP[k * 64 + j];
    float d2 = hh + sPn[k] - 2.0f * dot;
    d2 = fmaxf(d2, 0.0f);
    sim[k] = logf((d2 + 1.0f) / (d2 + 1e-4f));
  }

  float zacc = 0.0f;
#pragma unroll
  for (int j = 0; j < 8; ++j) {
    float z = sb0[j];
#pragma unroll
    for (int k = 0; k < 16; ++k) z += sim[k] * sW0[k * 8 + j];
    z = 0.5f * z * (1.0f + erff(z * 0.70710678118654752f));  // exact GELU
    zacc += z * sW1[j];
  }
  float val = zacc + sb1;
  out[i] = 1.0f / (1.0f + expf(-val));
  out[M + i] = (float)y[i];
}

// ---------------- host launcher ----------------

extern "C" void kernel_launch(void* const* d_in, const int* in_sizes, int n_in,
                              void* d_out, int out_size, void* d_ws, size_t ws_size,
                              hipStream_t stream) {
  (void)n_in; (void)out_size; (void)ws_size;

  const float* x    = (const float*)d_in[0];
  const int*   ei   = (const int*)d_in[1];
  const int*   yarr = (const int*)d_in[2];
  const float* W1   = (const float*)d_in[3];
  const float* b1   = (const float*)d_in[4];
  const float* W2   = (const float*)d_in[5];
  const float* b2   = (const float*)d_in[6];
  const float* W3   = (const float*)d_in[7];
  const float* b3   = (const float*)d_in[8];
  const float* prot = (const float*)d_in[9];
  const float* Wf0  = (const float*)d_in[10];
  const float* bf0  = (const float*)d_in[11];
  const float* Wf1  = (const float*)d_in[12];
  const float* bf1  = (const float*)d_in[13];
  float* out = (float*)d_out;

  const int M  = in_sizes[0] / 128;  // 50000 (== 3125 * 16)
  const int nE = in_sizes[1] / 2;    // 800000
  const int* src = ei;
  const int* dstp = ei + nE;

  // workspace carve: two ping-pong node buffers + degree/dinv (~52 MB)
  float* bufT = (float*)d_ws;                  // GEMM output (messages before aggregation)
  float* bufH = bufT + (size_t)M * 128;        // scatter accumulator / layer output
  float* deg  = bufH + (size_t)M * 128;
  float* dinv = deg + M;

  const int TB = 256;
  const int mtiles  = (M + 15) / 16;
  const int gblocks = (mtiles + 7) / 8;        // 8 waves per block
  const int eblocks = (nE + 7) / 8;            // 1 wave per edge

  deg_init_kernel<<<(M + TB - 1) / TB, TB, 0, stream>>>(deg, M);
  deg_acc_kernel<<<(nE + TB - 1) / TB, TB, 0, stream>>>(dstp, deg, nE);
  dinv_kernel<<<(M + TB - 1) / TB, TB, 0, stream>>>(deg, dinv, M);

  // Layer 1: 128 -> 128
  gcn_gemm_wmma<128, 128><<<gblocks, 256, 0, stream>>>(x, W1, bufT, M);
  hipMemsetAsync(bufH, 0, (size_t)M * 128 * sizeof(float), stream);
  edge_agg_kernel<<<eblocks, 256, 0, stream>>>(src, dstp, dinv, bufT, bufH, nE, 128);
  finish_kernel<<<((M * 128) + TB - 1) / TB, TB, 0, stream>>>(bufH, bufT, dinv, b1, M, 128);

  // Layer 2: 128 -> 64
  gcn_gemm_wmma<128, 64><<<gblocks, 256, 0, stream>>>(bufH, W2, bufT, M);
  hipMemsetAsync(bufH, 0, (size_t)M * 64 * sizeof(float), stream);
  edge_agg_kernel<<<eblocks, 256, 0, stream>>>(src, dstp, dinv, bufT, bufH, nE, 64);
  finish_kernel<<<((M * 64) + TB - 1) / TB, TB, 0, stream>>>(bufH, bufT, dinv, b2, M, 64);

  // Layer 3: 64 -> 64
  gcn_gemm_wmma<64, 64><<<gblocks, 256, 0, stream>>>(bufH, W3, bufT, M);
  hipMemsetAsync(bufH, 0, (size_t)M * 64 * sizeof(float), stream);
  edge_agg_kernel<<<eblocks, 256, 0, stream>>>(src, dstp, dinv, bufT, bufH, nE, 64);
  finish_kernel<<<((M * 64) + TB - 1) / TB, TB, 0, stream>>>(bufH, bufT, dinv, b3, M, 64);

  // Prototype distances + MLP readout + y cast
  readout_kernel<<<(M + TB - 1) / TB, TB, 0, stream>>>(bufH, prot, Wf0, bf0, Wf1, bf1,
                                                       yarr, out, M);
}